// SPINN_62259845923044
// MI455X (gfx1250) — compile-verified
//
#include <hip/hip_runtime.h>

typedef __attribute__((ext_vector_type(2))) float v2f;
typedef __attribute__((ext_vector_type(8))) float v8f;

#define DIM   3
#define NPTS  256
#define HID   64
#define RANK  64
#define FDIM  2
#define TOUT  (RANK * FDIM)   // 128

__device__ __forceinline__ float gelu_tanh(float x) {
    const float k0 = 0.7978845608028654f;  // sqrt(2/pi)
    const float k1 = 0.044715f;
    float x3 = x * x * x;
    return 0.5f * x * (1.0f + tanhf(k0 * (x + k1 * x3)));
}

// Issue one async memory->LDS 128-bit copy (GVS mode: SGPR64 base + VGPR32
// byte offset). Tracked by ASYNCcnt; publish with s_wait_asynccnt + barrier.
__device__ __forceinline__ void async_copy_b128(void* lds_dst, const float* gbase,
                                                unsigned byte_off) {
    unsigned lds_addr = (unsigned)(uintptr_t)lds_dst;   // low 32 bits = LDS offset
    asm volatile("global_load_async_to_lds_b128 %0, %1, %2"
                 :: "v"(lds_addr), "v"(byte_off), "s"((uint64_t)(uintptr_t)gbase)
                 : "memory");
}

__device__ __forceinline__ void async_wait_all() {
    asm volatile("s_wait_asynccnt 0x0" ::: "memory");
}

// -------------------------------------------------------------------------
// Kernel 1: per-dimension trunk MLPs -> tbuf[d][n][o], o = y*RANK + z (f32)
// One block per (dim, 16-point tile); layers 1..3 done with WMMA f32 16x16x4.
// -------------------------------------------------------------------------
__global__ __launch_bounds__(128) void spinn_trunk_kernel(
    const float* __restrict__ x,
    const float* __restrict__ W0, const float* __restrict__ b0,
    const float* __restrict__ W1, const float* __restrict__ b1,
    const float* __restrict__ W2, const float* __restrict__ b2,
    const float* __restrict__ W3, const float* __restrict__ b3,
    float* __restrict__ tbuf)
{
    __shared__ float hA[16][HID];
    __shared__ float hB[16][HID];

    const int d    = blockIdx.x >> 4;         // 0..2
    const int i0   = (blockIdx.x & 15) << 4;  // point-tile base

    const int tid  = threadIdx.x;
    const int lane = tid & 31;
    const int wave = tid >> 5;                // 0..3
    const int ln   = lane & 15;               // A row / B,C col
    const int koff = (lane >> 4) << 1;        // 0 or 2 (K sub-offset)
    const int mhi  = (lane >> 4) << 3;        // 0 or 8 (C row offset)

    // layer 0: h = gelu(x * W0 + b0)  (K = 1, scalar)
    for (int e = tid; e < 16 * HID; e += 128) {
        int p = e >> 6, hc = e & 63;
        float v = x[d * NPTS + i0 + p] * W0[d * HID + hc] + b0[d * HID + hc];
        hA[p][hc] = gelu_tanh(v);
    }
    __syncthreads();

    // layers 1 & 2: [16 x 64] @ W^T[64 x 64], gelu
    const float* Wl[2] = { W1 + d * HID * HID, W2 + d * HID * HID };
    const float* bl[2] = { b1 + d * HID,       b2 + d * HID       };
    #pragma unroll
    for (int l = 0; l < 2; ++l) {
        float (*src)[HID] = (l == 0) ? hA : hB;
        float (*dst)[HID] = (l == 0) ? hB : hA;
        const int col = (wave << 4) + ln;     // output column this lane owns
        v8f c = {};
        #pragma unroll
        for (int s = 0; s < 16; ++s) {
            const int z0 = (s << 2) + koff;
            v2f a, b;
            a.x = src[ln][z0];
            a.y = src[ln][z0 + 1];
            b.x = Wl[l][col * HID + z0];
            b.y = Wl[l][col * HID + z0 + 1];
            c = __builtin_amdgcn_wmma_f32_16x16x4_f32(false, a, false, b,
                                                      (short)0, c, false, false);
        }
        const float bb = bl[l][col];
        #pragma unroll
        for (int v = 0; v < 8; ++v)
            dst[v + mhi][col] = gelu_tanh(c[v] + bb);
        __syncthreads();
    }

    // layer 3: [16 x 64] @ W3^T[64 x 128], tanh, emit t
    for (int nt = wave; nt < 8; nt += 4) {
        const int col = (nt << 4) + ln;       // 0..127 output feature
        v8f c = {};
        #pragma unroll
        for (int s = 0; s < 16; ++s) {
            const int z0 = (s << 2) + koff;
            v2f a, b;
            a.x = hA[ln][z0];
            a.y = hA[ln][z0 + 1];
            b.x = W3[(d * TOUT + col) * HID + z0];
            b.y = W3[(d * TOUT + col) * HID + z0 + 1];
            c = __builtin_amdgcn_wmma_f32_16x16x4_f32(false, a, false, b,
                                                      (short)0, c, false, false);
        }
        const float bb = b3[d * TOUT + col];
        #pragma unroll
        for (int v = 0; v < 8; ++v) {
            int n = i0 + v + mhi;
            tbuf[(d * NPTS + n) * TOUT + col] = tanhf(c[v] + bb);
        }
    }
}

// -------------------------------------------------------------------------
// Kernel 2: CP contraction out[i,j,k,y] = sum_z t0[i,y,z]*t1[j,y,z]*t2[k,y,z]
// Block = (i-tile 16, k-tile 16, j-quarter of 64). t0/t2 tiles + t1 slab are
// staged into LDS with GLOBAL_LOAD_ASYNC_TO_LDS_B128 (ASYNCcnt). A fragments
// live in registers across the j loop; B is built per K-step as t1[j]*t2[col].
// 32 WMMAs per output tile; results leave via non-temporal float2 streaming
// stores (y pair adjacent) -> HBM-store-bound, which is this phase's roofline.
// -------------------------------------------------------------------------
__global__ __launch_bounds__(256) void spinn_cp_kernel(
    const float* __restrict__ tbuf, float* __restrict__ out)
{
    __shared__ float t0s[16 * TOUT];   //  8 KiB
    __shared__ float t2s[16 * TOUT];   //  8 KiB
    __shared__ float t1s[64 * TOUT];   // 32 KiB

    const int b     = blockIdx.x;
    const int it    = b >> 6;          // 0..15
    const int kt    = (b >> 2) & 15;   // 0..15
    const int jq    = b & 3;           // 0..3
    const int i0    = it << 4;
    const int k0    = kt << 4;
    const int jbase = jq << 6;

    const int tid  = threadIdx.x;
    const int lane = tid & 31;
    const int wave = tid >> 5;         // 0..7
    const int ln   = lane & 15;
    const int koff = (lane >> 4) << 1;
    const int mhi  = (lane >> 4) << 3;

    // Async stage: memory -> LDS directly (B128 per thread), no VGPR round-trip.
    const float* base0 = tbuf + (0 * NPTS + i0)    * TOUT;
    const float* base2 = tbuf + (2 * NPTS + k0)    * TOUT;
    const float* base1 = tbuf + (1 * NPTS + jbase) * TOUT;
    for (int e = tid * 4; e < 16 * TOUT; e += 256 * 4) {
        async_copy_b128(&t0s[e], base0, (unsigned)(e * 4));
        async_copy_b128(&t2s[e], base2, (unsigned)(e * 4));
    }
    for (int e = tid * 4; e < 64 * TOUT; e += 256 * 4) {
        async_copy_b128(&t1s[e], base1, (unsigned)(e * 4));
    }
    async_wait_all();
    __syncthreads();

    // A fragments for both field components, kept in registers for all j
    v2f a[FDIM][16];
    #pragma unroll
    for (int y = 0; y < FDIM; ++y) {
        #pragma unroll
        for (int s = 0; s < 16; ++s) {
            const int z0 = (s << 2) + koff;
            a[y][s].x = t0s[ln * TOUT + y * RANK + z0];
            a[y][s].y = t0s[ln * TOUT + y * RANK + z0 + 1];
        }
    }

    for (int jj = wave; jj < 64; jj += 8) {   // uniform per wave: EXEC all-1s
        const int j = jbase + jj;
        v8f c0 = {}, c1 = {};
        #pragma unroll
        for (int s = 0; s < 16; ++s) {
            const int z0 = (s << 2) + koff;
            v2f b0v, b1v;
            b0v.x = t1s[jj * TOUT + z0]            * t2s[ln * TOUT + z0];
            b0v.y = t1s[jj * TOUT + z0 + 1]        * t2s[ln * TOUT + z0 + 1];
            b1v.x = t1s[jj * TOUT + RANK + z0]     * t2s[ln * TOUT + RANK + z0];
            b1v.y = t1s[jj * TOUT + RANK + z0 + 1] * t2s[ln * TOUT + RANK + z0 + 1];
            c0 = __builtin_amdgcn_wmma_f32_16x16x4_f32(false, a[0][s], false, b0v,
                                                       (short)0, c0, false, false);
            c1 = __builtin_amdgcn_wmma_f32_16x16x4_f32(false, a[1][s], false, b1v,
                                                       (short)0, c1, false, false);
        }
        #pragma unroll
        for (int v = 0; v < 8; ++v) {
            const int i = i0 + v + mhi;
            const int k = k0 + ln;
            v2f pr;
            pr.x = c0[v];
            pr.y = c1[v];
            __builtin_nontemporal_store(
                pr, (v2f*)(out + (((size_t)i * NPTS + j) * NPTS + k) * FDIM));
        }
    }
}

extern "C" void kernel_launch(void* const* d_in, const int* in_sizes, int n_in,
                              void* d_out, int out_size, void* d_ws, size_t ws_size,
                              hipStream_t stream) {
    const float* x  = (const float*)d_in[0];
    const float* W0 = (const float*)d_in[1];
    const float* b0 = (const float*)d_in[2];
    const float* W1 = (const float*)d_in[3];
    const float* b1 = (const float*)d_in[4];
    const float* W2 = (const float*)d_in[5];
    const float* b2 = (const float*)d_in[6];
    const float* W3 = (const float*)d_in[7];
    const float* b3 = (const float*)d_in[8];
    float* out  = (float*)d_out;
    float* tbuf = (float*)d_ws;        // 3*256*128 f32 = 384 KiB scratch

    spinn_trunk_kernel<<<dim3(DIM * 16), dim3(128), 0, stream>>>(
        x, W0, b0, W1, b1, W2, b2, W3, b3, tbuf);
    spinn_cp_kernel<<<dim3(16 * 16 * 4), dim3(256), 0, stream>>>(tbuf, out);
}